// HungarianLoss_30133490549650
// MI455X (gfx1250) — compile-verified
//
#include <hip/hip_runtime.h>

// ---------------- problem constants (match reference) ----------------
#define W1f 2.0f
#define W2f 4.0f
#define W3f 1.0f
#define BSZ  64
#define NLN  32
#define DPAR 6
#define NY   410            // len(arange(0, 4096, 10))
#define BN   (BSZ * NLN)    // 2048 rows total
#define NYT  26             // ceil(410/16) column tiles
#define MTL  (BN / 16)      // 128 row tiles
#define BATCH_XY (NLN * NY) // 13120 floats per batch of px/gx

typedef __attribute__((ext_vector_type(2))) float        v2f;
typedef __attribute__((ext_vector_type(8))) float        v8f;
typedef __attribute__((ext_vector_type(4))) unsigned int u32x4;
typedef __attribute__((ext_vector_type(8))) int          i32x8;
typedef __attribute__((ext_vector_type(4))) int          i32x4;

#ifndef __has_builtin
#define __has_builtin(x) 0
#endif
#if __has_builtin(__builtin_amdgcn_tensor_load_to_lds) && __has_builtin(__builtin_amdgcn_s_wait_tensorcnt)
#define USE_TDM 1
#else
#define USE_TDM 0
#endif

// ================= Kernel 1: poly eval via V_WMMA_F32_16X16X4_F32 ==========
// px[m,n] = sum_k params[m,k] * y_n^k,  y_n = 10*n.  K=6 padded to 8 (2 steps).
// A 16x4 layout: lanes 0-15 / 16-31 both hold M=0..15; VGPR r holds K = 2*khalf + r.
// B 4x16 layout: VGPR r, lanes 0-15: K=r (N=lane); lanes 16-31: K=r+2 (N=lane-16).
// C/D 16x16:     VGPR r, lanes 0-15: M=r; lanes 16-31: M=r+8; N = lane&15.
__global__ __launch_bounds__(32) void polyeval_kernel(
    const float* __restrict__ pred_params, const float* __restrict__ gt_params,
    float* __restrict__ px, float* __restrict__ gx) {
  const int nt = blockIdx.x;   // 0..25
  const int mt = blockIdx.y;   // 0..127
  const int l  = threadIdx.x;  // 0..31
  const bool hi = (l >= 16);   // khalf
  const int nl = l & 15;
  const int m  = mt * 16 + nl; // A row this lane supplies
  const int n  = nt * 16 + nl; // output column this lane owns
  const float y  = 10.0f * (float)n;
  const float y2 = y * y;
  const float y3 = y2 * y;
  const float y4 = y2 * y2;
  const float y5 = y4 * y;

  v2f b0, b1;
  b0[0] = hi ? y2 : 1.0f;   // K = 2*khalf + 0
  b0[1] = hi ? y3 : y;      // K = 2*khalf + 1
  b1[0] = hi ? 0.f : y4;    // K = 4 + 2*khalf + 0 (K=6,7 rows are zero-padded)
  b1[1] = hi ? 0.f : y5;    // K = 4 + 2*khalf + 1

  const int e0 = hi ? 2 : 0;
  const float* pp = pred_params + m * DPAR;
  const float* gp = gt_params   + m * DPAR;
  v2f ap0, ap1, ag0, ag1;
  ap0[0] = pp[e0];     ap0[1] = pp[e0 + 1];
  ag0[0] = gp[e0];     ag0[1] = gp[e0 + 1];
  ap1[0] = hi ? 0.f : pp[4];  ap1[1] = hi ? 0.f : pp[5];
  ag1[0] = hi ? 0.f : gp[4];  ag1[1] = hi ? 0.f : gp[5];

  v8f cp, cg;
  for (int r = 0; r < 8; ++r) { cp[r] = 0.f; cg[r] = 0.f; }
  cp = __builtin_amdgcn_wmma_f32_16x16x4_f32(false, ap0, false, b0, (short)0, cp, false, false);
  cp = __builtin_amdgcn_wmma_f32_16x16x4_f32(false, ap1, false, b1, (short)0, cp, false, false);
  cg = __builtin_amdgcn_wmma_f32_16x16x4_f32(false, ag0, false, b0, (short)0, cg, false, false);
  cg = __builtin_amdgcn_wmma_f32_16x16x4_f32(false, ag1, false, b1, (short)0, cg, false, false);

  if (n < NY) {
    const int mbase = mt * 16 + (hi ? 8 : 0);
    for (int r = 0; r < 8; ++r) {
      px[(mbase + r) * NY + n] = cp[r];
      gx[(mbase + r) * NY + n] = cg[r];
    }
  }
}

// ================= Kernel 2: cost matrices (TDM -> LDS staging) ============
__global__ __launch_bounds__(256) void cost_kernel(
    const float* __restrict__ px, const float* __restrict__ gx,
    const float* __restrict__ pred_params, const float* __restrict__ gt_params,
    const float* __restrict__ pred_classes, const int* __restrict__ gt_classes,
    float* __restrict__ cost) {
  const int b = blockIdx.x;
  const int t = threadIdx.x;
  __shared__ __align__(16) float gx_s[BATCH_XY];
  const float* gsrc = gx + (long long)b * BATCH_XY;

  __builtin_prefetch(px + (long long)b * BATCH_XY + t * 16, 0, 1);

#if USE_TDM
  if (t < 32) {  // wave 0 issues one TDM op for the whole 52.5 KB tile
    unsigned long long ga = (unsigned long long)(const void*)gsrc;
    unsigned int lds_off  = (unsigned int)(unsigned long long)(void*)&gx_s[0];
    u32x4 g0;
    g0[0] = 1u;                                              // count=1
    g0[1] = lds_off;                                         // lds_addr
    g0[2] = (unsigned int)(ga & 0xFFFFFFFFull);              // global_addr lo
    g0[3] = (unsigned int)((ga >> 32) & 0x1FFFFFFull) | (2u << 30); // addr hi | type=2
    i32x8 g1;
    g1[0] = (int)(2u << 16);                    // data_size = 4B
    g1[1] = (int)((unsigned)BATCH_XY << 16);    // tensor_dim0[15:0]
    g1[2] = (int)(1u << 16);                    // tensor_dim0 hi=0, tensor_dim1=1
    g1[3] = (int)((unsigned)BATCH_XY << 16);    // tile_dim0 = 13120
    g1[4] = 1;                                  // tile_dim1 = 1
    g1[5] = BATCH_XY;                           // tensor_dim0_stride
    g1[6] = 0;
    g1[7] = 0;
    i32x4 gz; gz[0] = 0; gz[1] = 0; gz[2] = 0; gz[3] = 0;
    i32x8 gz8; for (int k = 0; k < 8; ++k) gz8[k] = 0;
    // 6-arg form (this toolchain): (g0, g1, g2, g3, g4, cpol)
    __builtin_amdgcn_tensor_load_to_lds(g0, g1, gz, gz, gz8, 0);
    __builtin_amdgcn_s_wait_tensorcnt(0);
  }
#else
  {
    const float4* s4 = (const float4*)gsrc;
    float4* d4 = (float4*)gx_s;
    for (int i = t; i < BATCH_XY / 4; i += 256) d4[i] = s4[i];
  }
#endif
  __syncthreads();

  const int p  = t & 31;         // pred column (same for this thread's 4 pairs)
  const int g0 = t >> 5;         // first gt row
  const float* pxr = px + (long long)(b * NLN + p) * NY;
  const float* pp  = pred_params + (b * NLN + p) * DPAR;
  for (int gi = 0; gi < 4; ++gi) {
    const int g = g0 + 8 * gi;
    const float* gxr = gx_s + g * NY;
    float path = 0.f;
    for (int y = 0; y < NY; ++y) path += fabsf(gxr[y] - pxr[y]);
    float parm = 0.f;
    const float* gp = gt_params + (b * NLN + g) * DPAR;
    for (int d = 0; d < DPAR; ++d) parm += fabsf(gp[d] - pp[d]);
    const int gc = gt_classes[b * NLN + g];
    const float cls = -pred_classes[(b * NLN + p) * 2 + gc];
    cost[b * NLN * NLN + g * NLN + p] =
        W1f * cls + (float)gc * (W2f * path + W3f * parm);
  }
}

// ================= Kernel 3: Hungarian (reference-exact semantics) =========
// One wave per batch; lane owns column j=lane+1; shuffle argmin (tie->smallest j).
// Mirrors the reference numpy exactly: minv/way copies are never persisted, so
// each iteration's candidate is just cur, and the "augment" is p[j0_final]=i.
__global__ __launch_bounds__(32) void hungarian_kernel(
    const float* __restrict__ cost, int* __restrict__ perm) {
  const int b    = blockIdx.x;
  const int lane = threadIdx.x;
  const int j    = lane + 1;
  __shared__ float cost_s[NLN * NLN];
  __shared__ float u_s[NLN + 1];
  __shared__ int   p_s[NLN + 1];
  for (int idx = lane; idx < NLN * NLN; idx += 32)
    cost_s[idx] = cost[b * NLN * NLN + idx];
  u_s[lane] = 0.0f; p_s[lane] = 0;
  if (lane == 0) { u_s[NLN] = 0.0f; p_s[NLN] = 0; }
  __syncthreads();

  const float INF = 3.0e38f;
  float v_j = 0.0f;
  for (int i = 1; i <= NLN; ++i) {
    if (lane == 0) p_s[0] = i;
    __syncthreads();
    int  j0     = 0;
    bool used_j = false;
    bool used0  = false;
    for (;;) {
      if (j0 == 0) used0 = true;
      if (j == j0) used_j = true;
      const int   i0  = p_s[j0];
      const float cur = cost_s[(i0 - 1) * NLN + lane] - u_s[i0] - v_j;
      float bv = used_j ? INF : cur;
      int   bj = j;
      for (int off = 16; off > 0; off >>= 1) {
        const float ov = __shfl_xor(bv, off, 32);
        const int   oj = __shfl_xor(bj, off, 32);
        if (ov < bv || (ov == bv && oj < bj)) { bv = ov; bj = oj; }
      }
      const float delta = bv;
      const int   j1    = bj;
      __syncthreads();
      if (used_j) { u_s[p_s[j]] += delta; v_j -= delta; }
      if (lane == 0 && used0) { u_s[p_s[0]] += delta; }
      __syncthreads();
      j0 = j1;
      if (p_s[j0] == 0) break;
    }
    if (lane == 0) p_s[j0] = i;
    __syncthreads();
  }
  perm[b * NLN + (p_s[j] - 1)] = j - 1;  // row_to_col
}

// ================= Kernel 4: per-batch loss partials =======================
__global__ __launch_bounds__(256) void loss_partial_kernel(
    const float* __restrict__ px, const float* __restrict__ gx,
    const float* __restrict__ pred_params, const float* __restrict__ gt_params,
    const float* __restrict__ pred_classes, const int* __restrict__ gt_classes,
    const int* __restrict__ perm, float* __restrict__ partials) {
  const int b = blockIdx.x;
  const int t = threadIdx.x;
  const int g = t >> 3;   // 0..31
  const int s = t & 7;    // y-slice
  const int pr = perm[b * NLN + g];
  const int gc = gt_classes[b * NLN + g];
  const float* pxr = px + (long long)(b * NLN + pr) * NY;
  const float* gxr = gx + (long long)(b * NLN + g) * NY;
  float acc = 0.f;
  for (int y = s; y < NY; y += 8) acc += fabsf(pxr[y] - gxr[y]);
  // y-coords of pred_pts/gt_pts are identical -> only x contributes; mean over (410,2)
  float term = (gc == 1) ? acc * (W2f / (2.0f * (float)NY)) : 0.0f;
  if (s == 0) {
    float parm = 0.f;
    const float* pp = pred_params + (b * NLN + pr) * DPAR;
    const float* gp = gt_params   + (b * NLN + g)  * DPAR;
    for (int d = 0; d < DPAR; ++d) parm += fabsf(pp[d] - gp[d]);
    const float cls = pred_classes[(b * NLN + pr) * 2 + gc];
    term += -W1f * cls + ((gc == 1) ? parm * (W3f / (float)DPAR) : 0.0f);
  }
  __shared__ float red[256];
  red[t] = term;
  __syncthreads();
  for (int off = 128; off > 0; off >>= 1) {
    if (t < off) red[t] += red[t + off];
    __syncthreads();
  }
  if (t == 0) partials[b] = red[0];
}

// ================= Kernel 5: deterministic final reduction =================
__global__ __launch_bounds__(64) void finalize_kernel(
    const float* __restrict__ partials, float* __restrict__ out) {
  const int t = threadIdx.x;
  __shared__ float red[64];
  red[t] = partials[t];
  __syncthreads();
  for (int off = 32; off > 0; off >>= 1) {
    if (t < off) red[t] += red[t + off];
    __syncthreads();
  }
  if (t == 0) out[0] = red[0] / (float)BSZ;
}

// ================= host-side launcher =====================================
extern "C" void kernel_launch(void* const* d_in, const int* in_sizes, int n_in,
                              void* d_out, int out_size, void* d_ws, size_t ws_size,
                              hipStream_t stream) {
  const float* pred_params  = (const float*)d_in[0];
  const float* pred_classes = (const float*)d_in[1];
  const float* gt_params    = (const float*)d_in[2];
  const int*   gt_classes   = (const int*)d_in[3];
  (void)in_sizes; (void)n_in; (void)out_size; (void)ws_size;

  float* px       = (float*)d_ws;             // 2048*410
  float* gx       = px + (size_t)BN * NY;     // 2048*410
  float* cost     = gx + (size_t)BN * NY;     // 64*32*32
  int*   perm     = (int*)(cost + BSZ * NLN * NLN); // 64*32
  float* partials = (float*)(perm + BSZ * NLN);     // 64

  polyeval_kernel<<<dim3(NYT, MTL), 32, 0, stream>>>(pred_params, gt_params, px, gx);
  cost_kernel<<<BSZ, 256, 0, stream>>>(px, gx, pred_params, gt_params,
                                       pred_classes, gt_classes, cost);
  hungarian_kernel<<<BSZ, 32, 0, stream>>>(cost, perm);
  loss_partial_kernel<<<BSZ, 256, 0, stream>>>(px, gx, pred_params, gt_params,
                                               pred_classes, gt_classes, perm, partials);
  finalize_kernel<<<1, 64, 0, stream>>>(partials, (float*)d_out);
}